// PiecewiseSparseMLP_51737176047981
// MI455X (gfx1250) — compile-verified
//
#include <hip/hip_runtime.h>

typedef float v2f __attribute__((ext_vector_type(2)));
typedef float v4f __attribute__((ext_vector_type(4)));
typedef float v8f __attribute__((ext_vector_type(8)));

#define K_EXP 32
#define D_IN  10
#define D_INP 12    // D_IN padded: cols 0..9 = x, col 10 = 1.0 (bias), col 11 = 0
#define D_H   20
#define D_HP  32    // hidden padded to 2 N-tiles of 16
#define WAVES_PER_BLOCK 8
#define ROWS_PER_WAVE   16
#define ROWS_PER_BLOCK  (WAVES_PER_BLOCK * ROWS_PER_WAVE)

__device__ __forceinline__ float relu1(float x) {
    // single v_med3_f32: med3(x, 0, FLT_MAX) == clamp(x, 0, FLT_MAX) == relu
    return __builtin_amdgcn_fmed3f(x, 0.0f, 3.402823466e38f);
}

__global__ __launch_bounds__(256) void
moe_wmma_kernel(const float* __restrict__ X,  const float* __restrict__ W1,
                const float* __restrict__ B1, const float* __restrict__ W2,
                const float* __restrict__ B2, const float* __restrict__ P,
                float* __restrict__ Out)
{
    __shared__ __align__(16) float sW1[K_EXP][D_HP][D_INP];                    // 48 KB (bias in col 10)
    __shared__ __align__(16) float sW2[K_EXP][D_HP];                           //  4 KB
    __shared__ __align__(16) float sB2[K_EXP];                                 // 128 B
    __shared__ __align__(16) float sP [K_EXP][D_INP];                          // 1.5 KB
    __shared__ __align__(16) float sX [WAVES_PER_BLOCK][ROWS_PER_WAVE][D_INP]; //  6 KB
    __shared__ __align__(16) float sG [WAVES_PER_BLOCK][K_EXP][ROWS_PER_WAVE]; // 16 KB

    const int tid  = threadIdx.x;
    const int lane = tid & 31;
    const int wv   = tid >> 5;
    const int hi   = lane >> 4;   // 0 or 1 (lane half)
    const int lo   = lane & 15;

    // ---------------- phase 0: zero padded shared regions ----------------
    {
        float* z = &sW1[0][0][0];
        for (int i = tid; i < K_EXP * D_HP * D_INP; i += 256) z[i] = 0.f;
        z = &sW2[0][0];
        for (int i = tid; i < K_EXP * D_HP; i += 256) z[i] = 0.f;
        z = &sP[0][0];
        for (int i = tid; i < K_EXP * D_INP; i += 256) z[i] = 0.f;
    }
    __syncthreads();

    // ---------------- phase 1: stage weights + X tile ----------------
    for (int e = tid; e < K_EXP * D_H * D_IN; e += 256) {
        int k = e / (D_H * D_IN);
        int r = e - k * (D_H * D_IN);
        int h = r / D_IN;
        int d = r - h * D_IN;
        sW1[k][h][d] = W1[e];
    }
    for (int e = tid; e < K_EXP * D_H; e += 256) {
        int k = e / D_H;
        int h = e - k * D_H;
        sW1[k][h][10] = B1[e];   // bias folded into the augmented input column
        sW2[k][h]     = W2[e];   // W2 is [K,1,20] -> flat k*20+h
    }
    if (tid < K_EXP) sB2[tid] = B2[tid];
    for (int e = tid; e < K_EXP * D_IN; e += 256) {
        int k = e / D_IN;
        int d = e - k * D_IN;
        sP[k][d] = P[e];
    }

    const int row0 = blockIdx.x * ROWS_PER_BLOCK + wv * ROWS_PER_WAVE;
    {   // coalesced copy of 16 rows x 10 floats into padded LDS tile
        const float* xrow = X + (size_t)row0 * D_IN;
        #pragma unroll
        for (int i = 0; i < 5; ++i) {
            int f = lane + i * 32;           // 0..159
            int r = f / D_IN;
            int c = f - r * D_IN;
            sX[wv][r][c] = xrow[f];
        }
        if (hi == 0) { sX[wv][lo][10] = 1.0f; sX[wv][lo][11] = 0.f; }  // bias column = 1
    }
    __syncthreads();

    // ---------------- phase 2: gating (exp(-dist) + b2 fold) ----------------
    float xr[D_IN];
    #pragma unroll
    for (int d = 0; d < D_IN; ++d) xr[d] = sX[wv][lo][d];

    float psum = 0.f;    // sum_k g[m,k]        (softmax denominator)
    float bpart = 0.f;   // sum_k g[m,k]*b2[k]  (bias term of weighted preds)
    #pragma unroll
    for (int kk = 0; kk < 16; ++kk) {
        int k = hi * 16 + kk;
        float d2 = 0.f;
        #pragma unroll
        for (int d = 0; d < D_IN; ++d) {
            float df = xr[d] - sP[k][d];
            d2 = fmaf(df, df, d2);
        }
        float g = __expf(-sqrtf(d2));
        sG[wv][k][lo] = g;                 // transposed: [expert][row]
        psum  += g;
        bpart  = fmaf(g, sB2[k], bpart);
    }
    float tot  = psum  + __shfl_xor(psum,  16, 32);  // full denom for row `lo`
    float bsum = bpart + __shfl_xor(bpart, 16, 32);  // full b2 term for row `lo`
    float inv  = 1.0f / tot;

    // after the final reduction, this lane holds the result for this row:
    const int row_out = hi * 8 + ((lane >> 1) & 7);
    float invd  = __shfl(inv,  row_out, 32);   // lanes 0..15 hold values for row==lane
    float bsumd = __shfl(bsum, row_out, 32);

    // ---------------- phase 3: A matrix (reused across all experts) ----------------
    // 16x4 f32 A layout: V0 lane L = Xaug[L&15, k0 + 2*(L>>4)], V1 = ... + 1
    const float* xp = &sX[wv][lo][2 * hi];
    v2f A0 = *(const v2f*)(xp + 0);
    v2f A1 = *(const v2f*)(xp + 4);
    v2f A2 = *(const v2f*)(xp + 8);

    // ---------------- phase 4: expert loop (no per-expert reduction) ----------------
    const v8f zeroC = {0.f, 0.f, 0.f, 0.f, 0.f, 0.f, 0.f, 0.f};
    float R[8] = {0.f, 0.f, 0.f, 0.f, 0.f, 0.f, 0.f, 0.f};

    for (int k = 0; k < K_EXP; ++k) {
        // B = W1aug^T tiles: B[kk][n] = W1aug[k][ntile*16+n][kk]
        const float* b0  = &sW1[k][lo][2 * hi];
        const float* b1p = &sW1[k][16 + lo][2 * hi];
        v2f B00 = *(const v2f*)(b0 + 0);
        v2f B01 = *(const v2f*)(b0 + 4);
        v2f B02 = *(const v2f*)(b0 + 8);
        v2f B10 = *(const v2f*)(b1p + 0);
        v2f B11 = *(const v2f*)(b1p + 4);
        v2f B12 = *(const v2f*)(b1p + 8);

        float wlo = sW2[k][lo];
        float whi = sW2[k][16 + lo];

        // gate vector for this expert: rows hi*8 .. hi*8+7
        v4f gA = *(const v4f*)&sG[wv][k][hi * 8];
        v4f gB = *(const v4f*)&sG[wv][k][hi * 8 + 4];

        v8f acc0 = __builtin_amdgcn_wmma_f32_16x16x4_f32(false, A0, false, B00, (short)0, zeroC, false, false);
        acc0     = __builtin_amdgcn_wmma_f32_16x16x4_f32(false, A1, false, B01, (short)0, acc0,  false, false);
        acc0     = __builtin_amdgcn_wmma_f32_16x16x4_f32(false, A2, false, B02, (short)0, acc0,  false, false);
        v8f acc1 = __builtin_amdgcn_wmma_f32_16x16x4_f32(false, A0, false, B10, (short)0, zeroC, false, false);
        acc1     = __builtin_amdgcn_wmma_f32_16x16x4_f32(false, A1, false, B11, (short)0, acc1,  false, false);
        acc1     = __builtin_amdgcn_wmma_f32_16x16x4_f32(false, A2, false, B12, (short)0, acc1,  false, false);

        #pragma unroll
        for (int r = 0; r < 8; ++r) {
            float g  = (r < 4) ? gA[r] : gB[r - 4];
            float h0 = relu1(acc0[r]);                 // bias already inside (aug column)
            float h1 = relu1(acc1[r]);
            float s  = fmaf(h0, wlo, h1 * whi);        // per-lane (n) partial of pred
            R[r] = fmaf(s, g, R[r]);                   // gate-weighted accumulate over experts
        }
    }

    // ---------------- phase 5: single select-butterfly reduction ----------------
    // reduce 8 regs x 16 lanes -> lane L holds row (L>>1)&7 (+ half*8)
    const bool s3 = (lane & 8) != 0;
    float u[4];
    #pragma unroll
    for (int r = 0; r < 4; ++r) {
        float a = R[r]     + __shfl_xor(R[r],     8, 32);
        float b = R[r + 4] + __shfl_xor(R[r + 4], 8, 32);
        u[r] = s3 ? b : a;
    }
    const bool s2 = (lane & 4) != 0;
    float v[2];
    #pragma unroll
    for (int r = 0; r < 2; ++r) {
        float a = u[r]     + __shfl_xor(u[r],     4, 32);
        float b = u[r + 2] + __shfl_xor(u[r + 2], 4, 32);
        v[r] = s2 ? b : a;
    }
    const bool s1 = (lane & 2) != 0;
    float a2  = v[0] + __shfl_xor(v[0], 2, 32);
    float b2v = v[1] + __shfl_xor(v[1], 2, 32);
    float wred = s1 ? b2v : a2;
    float red  = wred + __shfl_xor(wred, 1, 32);   // sum over hidden lanes & experts

    float outv = (red + bsumd) * invd;             // add b2 term, normalize softmax
    if ((lane & 1) == 0) Out[row0 + row_out] = outv;
}

extern "C" void kernel_launch(void* const* d_in, const int* in_sizes, int n_in,
                              void* d_out, int out_size, void* d_ws, size_t ws_size,
                              hipStream_t stream)
{
    const float* X  = (const float*)d_in[0];
    const float* W1 = (const float*)d_in[1];
    const float* B1 = (const float*)d_in[2];
    const float* W2 = (const float*)d_in[3];
    const float* B2 = (const float*)d_in[4];
    const float* P  = (const float*)d_in[5];
    float* Out = (float*)d_out;

    const int Bt = in_sizes[0] / D_IN;         // 262144
    const int blocks = Bt / ROWS_PER_BLOCK;    // exact: 2048
    hipLaunchKernelGGL(moe_wmma_kernel, dim3(blocks), dim3(256), 0, stream,
                       X, W1, B1, W2, B2, P, Out);
}